// cQSSM_27582279975484
// MI455X (gfx1250) — compile-verified
//
#include <hip/hip_runtime.h>
#include <math.h>

#define HID     64
#define INP     128
#define CATW    192      // INPUT + HIDDEN
#define NQ      4
#define TSTEPS  512
#define ROWS    16       // batch rows per block
#define NTHR    128      // 4 waves

typedef float v2f __attribute__((ext_vector_type(2)));
typedef float v8f __attribute__((ext_vector_type(8)));

__device__ __forceinline__ float sigmf(float x) { return 1.0f / (1.0f + expf(-x)); }

// ---- 4-qubit gate helpers. Lane j (0..7) of a row holds amplitudes j (s0) and j+8 (s1).
// qubit 0 <-> intra-lane (bit 8); qubit q>0 <-> shfl_xor mask (8>>q) in {4,2,1}.

__device__ __forceinline__ void gate_h(int q, int j,
    float& s0r, float& s0i, float& s1r, float& s1i) {
  const float K = 0.70710678118654752f;
  if (q == 0) {
    float t0r = s0r, t0i = s0i;
    s0r = K * (t0r + s1r); s0i = K * (t0i + s1i);
    s1r = K * (t0r - s1r); s1i = K * (t0i - s1i);
  } else {
    int m = 8 >> q;
    float sg = (j & m) ? -1.0f : 1.0f;
    float p0r = __shfl_xor(s0r, m, 32), p0i = __shfl_xor(s0i, m, 32);
    float p1r = __shfl_xor(s1r, m, 32), p1i = __shfl_xor(s1i, m, 32);
    s0r = K * (sg * s0r + p0r); s0i = K * (sg * s0i + p0i);
    s1r = K * (sg * s1r + p1r); s1i = K * (sg * s1i + p1i);
  }
}

__device__ __forceinline__ void gate_ry(int q, int j, float c, float s,
    float& s0r, float& s0i, float& s1r, float& s1i) {
  if (q == 0) {
    float t0r = s0r, t0i = s0i;
    s0r = c * t0r - s * s1r; s0i = c * t0i - s * s1i;
    s1r = s * t0r + c * s1r; s1i = s * t0i + c * s1i;
  } else {
    int m = 8 >> q;
    float st = (j & m) ? s : -s;
    float p0r = __shfl_xor(s0r, m, 32), p0i = __shfl_xor(s0i, m, 32);
    float p1r = __shfl_xor(s1r, m, 32), p1i = __shfl_xor(s1i, m, 32);
    s0r = c * s0r + st * p0r; s0i = c * s0i + st * p0i;
    s1r = c * s1r + st * p1r; s1i = c * s1i + st * p1i;
  }
}

__device__ __forceinline__ void gate_rz(int q, int j, float c, float s,
    float& s0r, float& s0i, float& s1r, float& s1i) {
  if (q == 0) {
    // s0 *= (c - i s), s1 *= (c + i s)
    float a = s0r, b = s0i;
    s0r = a * c + b * s;  s0i = b * c - a * s;
    a = s1r; b = s1i;
    s1r = a * c - b * s;  s1i = b * c + a * s;
  } else {
    int m = 8 >> q;
    float ss = (j & m) ? s : -s;   // both amps share the same bit for q>0
    float a = s0r, b = s0i;
    s0r = a * c - b * ss; s0i = b * c + a * ss;
    a = s1r; b = s1i;
    s1r = a * c - b * ss; s1i = b * c + a * ss;
  }
}

__device__ __forceinline__ void gate_rx(int q, int j, float c, float s,
    float& s0r, float& s0i, float& s1r, float& s1i) {
  if (q == 0) {
    float t0r = s0r, t0i = s0i;
    s0r = c * t0r + s * s1i; s0i = c * t0i - s * s1r;
    s1r = c * s1r + s * t0i; s1i = c * s1i - s * t0r;
  } else {
    int m = 8 >> q;
    float p0r = __shfl_xor(s0r, m, 32), p0i = __shfl_xor(s0i, m, 32);
    float p1r = __shfl_xor(s1r, m, 32), p1i = __shfl_xor(s1i, m, 32);
    s0r = c * s0r + s * p0i; s0i = c * s0i - s * p0r;
    s1r = c * s1r + s * p1i; s1i = c * s1i - s * p1r;
  }
}

__global__ __launch_bounds__(NTHR)
void cqssm_persistent_kernel(const float* __restrict__ x,
                             const float* __restrict__ fw,
                             const float* __restrict__ entry_w,
                             const float* __restrict__ entry_b,
                             const float* __restrict__ exit_w,
                             const float* __restrict__ exit_b,
                             const float* __restrict__ vqc_w,
                             const float* __restrict__ norm_w,
                             const float* __restrict__ norm_b,
                             float* __restrict__ out,
                             int Btotal) {
  __shared__ float  lds_cat[ROWS][CATW];    // [x_t (scaled) | h]
  __shared__ float  lds_out[ROWS][CATW];    // exit GEMM result
  __shared__ float  lds_ypart[4][ROWS][NQ]; // per-wave partial y
  __shared__ float  lds_y[ROWS][NQ];
  __shared__ float  lds_yq[ROWS][NQ];       // y + q (exit A operand)
  __shared__ float4 lds_trig[ROWS][NQ];     // cy, sy, cz, sz
  __shared__ float2 lds_wtrig[24];          // cos/sin of vqc_weights/2
  __shared__ float  lds_expfw[INP];

  const int tid  = threadIdx.x;
  const int lane = tid & 31;
  const int wave = tid >> 5;
  const int row0 = blockIdx.x * ROWS;

  // ---------------- one-time setup ----------------
  if (tid < INP) lds_expfw[tid] = expf(fw[tid]);
  if (tid < 24) {
    float w = vqc_w[tid];
    lds_wtrig[tid] = make_float2(cosf(0.5f * w), sinf(0.5f * w));
  }

  // elementwise mapping: 16 rows x 8 lanes; lane ej covers hdims ej, ej+8, ... ej+56
  const int er = tid >> 3;
  const int ej = tid & 7;
  float c_reg[8], nrmw[8], nrmb[8];
  #pragma unroll
  for (int i = 0; i < 8; ++i) {
    c_reg[i] = 0.0f;
    lds_cat[er][INP + ej + 8 * i] = 0.0f;   // h0 = 0
    nrmw[i] = norm_w[ej + 8 * i];
    nrmb[i] = norm_b[ej + 8 * i];
  }

  // WMMA operand constants (wave32 16x16x4 f32 layouts)
  const int bn = lane & 15;            // N column (B/C/D), or M row (A)
  const int k0 = (lane >> 4) * 2;      // K sub-pair selected by lane half
  const int m0 = (lane >> 4) * 8;      // C/D row base per lane half

  // preload entry-GEMM B operands: 12 K-chunks per wave; cols >= NQ are zero
  v2f eB[12];
  #pragma unroll
  for (int cc = 0; cc < 12; ++cc) {
    int kc = (wave * 12 + cc) * 4 + k0;
    if (bn < NQ) { eB[cc][0] = entry_w[bn * CATW + kc];
                   eB[cc][1] = entry_w[bn * CATW + kc + 1]; }
    else         { eB[cc][0] = 0.0f; eB[cc][1] = 0.0f; }
  }
  // preload exit-GEMM B operands + bias: wave handles N-tiles {wave, wave+4, wave+8}
  v2f  xB[3];
  float xbias[3];
  #pragma unroll
  for (int tix = 0; tix < 3; ++tix) {
    int n = (wave + 4 * tix) * 16 + bn;
    xB[tix][0] = exit_w[n * NQ + k0];
    xB[tix][1] = exit_w[n * NQ + k0 + 1];
    xbias[tix] = exit_b[n];
  }
  float eb_r = (tid < ROWS * NQ) ? entry_b[tid & 3] : 0.0f;

  __syncthreads();

  // ---------------- time loop ----------------
  for (int t = 0; t < TSTEPS; ++t) {
    // Phase 1: stage x_t * exp(fw) into LDS (16 rows x 128), prefetch next tile
    {
      int r = tid >> 3, j = tid & 7;
      const float4* xr = (const float4*)(x + ((size_t)(row0 + r) * TSTEPS + t) * INP);
      #pragma unroll
      for (int i = 0; i < 4; ++i) {
        int c4 = j + 8 * i;
        float4 v = xr[c4];
        v.x *= lds_expfw[c4 * 4 + 0]; v.y *= lds_expfw[c4 * 4 + 1];
        v.z *= lds_expfw[c4 * 4 + 2]; v.w *= lds_expfw[c4 * 4 + 3];
        ((float4*)&lds_cat[r][0])[c4] = v;
      }
      if (t + 1 < TSTEPS)
        __builtin_prefetch(x + ((size_t)(row0 + r) * TSTEPS + (t + 1)) * INP + j * 16, 0, 3);
    }
    __syncthreads();

    // Phase 2: entry GEMM  y[16,4] = cat[16,192] @ entry_w^T  via 12 wmma K-chunks/wave
    {
      v8f acc = {};
      #pragma unroll
      for (int cc = 0; cc < 12; ++cc) {
        int kc = (wave * 12 + cc) * 4 + k0;
        v2f a;
        a[0] = lds_cat[bn][kc];
        a[1] = lds_cat[bn][kc + 1];
        acc = __builtin_amdgcn_wmma_f32_16x16x4_f32(
            false, a, false, eB[cc], (short)0, acc, false, false);
      }
      if (bn < NQ) {
        #pragma unroll
        for (int r8 = 0; r8 < 8; ++r8) lds_ypart[wave][m0 + r8][bn] = acc[r8];
      }
    }
    __syncthreads();

    // Phase 3: reduce partial y across waves, add bias, compute encoding trig
    if (tid < ROWS * NQ) {
      int r = tid >> 2, n = tid & 3;
      float y = eb_r;
      #pragma unroll
      for (int w = 0; w < 4; ++w) y += lds_ypart[w][r][n];
      lds_y[r][n] = y;
      float ty = atanf(y), tz = atanf(y * y);
      lds_trig[r][n] = make_float4(cosf(0.5f * ty), sinf(0.5f * ty),
                                   cosf(0.5f * tz), sinf(0.5f * tz));
    }
    __syncthreads();

    // Phase 4: VQC — 8 lanes per row, 2 complex amplitudes per lane
    {
      int r = tid >> 3, j = tid & 7;
      float s0r = (j == 0) ? 1.0f : 0.0f, s0i = 0.0f, s1r = 0.0f, s1i = 0.0f;

      #pragma unroll
      for (int q = 0; q < NQ; ++q) {        // encoding: H, RY(atan y), RZ(atan y^2)
        float4 tg = lds_trig[r][q];
        gate_h (q, j, s0r, s0i, s1r, s1i);
        gate_ry(q, j, tg.x, tg.y, s0r, s0i, s1r, s1i);
        gate_rz(q, j, tg.z, tg.w, s0r, s0i, s1r, s1i);
      }
      #pragma unroll
      for (int l = 0; l < 2; ++l) {
        // CNOT(0,1): ctrl = bit8 (s1), tgt = ^4
        { float a = __shfl_xor(s1r, 4, 32), b = __shfl_xor(s1i, 4, 32); s1r = a; s1i = b; }
        // CNOT(1,2): ctrl = j&4, tgt = ^2
        { float a0 = __shfl_xor(s0r, 2, 32), b0 = __shfl_xor(s0i, 2, 32);
          float a1 = __shfl_xor(s1r, 2, 32), b1 = __shfl_xor(s1i, 2, 32);
          if (j & 4) { s0r = a0; s0i = b0; s1r = a1; s1i = b1; } }
        // CNOT(2,3): ctrl = j&2, tgt = ^1
        { float a0 = __shfl_xor(s0r, 1, 32), b0 = __shfl_xor(s0i, 1, 32);
          float a1 = __shfl_xor(s1r, 1, 32), b1 = __shfl_xor(s1i, 1, 32);
          if (j & 2) { s0r = a0; s0i = b0; s1r = a1; s1i = b1; } }
        // CNOT(3,0): ctrl = j&1, tgt = bit8 (swap s0<->s1)
        { if (j & 1) { float a = s0r, b = s0i; s0r = s1r; s0i = s1i; s1r = a; s1i = b; } }

        #pragma unroll
        for (int q = 0; q < NQ; ++q) {
          float2 wx = lds_wtrig[(l * 3 + 0) * 4 + q];
          float2 wy = lds_wtrig[(l * 3 + 1) * 4 + q];
          float2 wz = lds_wtrig[(l * 3 + 2) * 4 + q];
          gate_rx(q, j, wx.x, wx.y, s0r, s0i, s1r, s1i);
          gate_ry(q, j, wy.x, wy.y, s0r, s0i, s1r, s1i);
          gate_rz(q, j, wz.x, wz.y, s0r, s0i, s1r, s1i);
        }
      }
      // Z expectation values
      float p0 = s0r * s0r + s0i * s0i, p1 = s1r * s1r + s1i * s1i;
      float tot = p0 + p1;
      float e0 = p0 - p1;
      float e1 = (j & 4) ? -tot : tot;
      float e2 = (j & 2) ? -tot : tot;
      float e3 = (j & 1) ? -tot : tot;
      #pragma unroll
      for (int m = 1; m < 8; m <<= 1) {
        e0 += __shfl_xor(e0, m, 32);
        e1 += __shfl_xor(e1, m, 32);
        e2 += __shfl_xor(e2, m, 32);
        e3 += __shfl_xor(e3, m, 32);
      }
      if (j == 0) {
        lds_yq[r][0] = lds_y[r][0] + e0;
        lds_yq[r][1] = lds_y[r][1] + e1;
        lds_yq[r][2] = lds_y[r][2] + e2;
        lds_yq[r][3] = lds_y[r][3] + e3;
      }
    }
    __syncthreads();

    // Phase 5: exit GEMM  out[16,192] = (y+q)[16,4] @ exit_w^T  — one wmma per tile
    {
      v2f a;
      a[0] = lds_yq[bn][k0];
      a[1] = lds_yq[bn][k0 + 1];
      #pragma unroll
      for (int tix = 0; tix < 3; ++tix) {
        v8f cacc;
        #pragma unroll
        for (int r8 = 0; r8 < 8; ++r8) cacc[r8] = xbias[tix];   // fold exit_b into C
        v8f d = __builtin_amdgcn_wmma_f32_16x16x4_f32(
            false, a, false, xB[tix], (short)0, cacc, false, false);
        int ncol = (wave + 4 * tix) * 16 + bn;
        #pragma unroll
        for (int r8 = 0; r8 < 8; ++r8) lds_out[m0 + r8][ncol] = d[r8];
      }
    }
    __syncthreads();

    // Phase 6: gates, cell update, layernorm, emit h
    {
      float v[8], sum = 0.0f, sumsq = 0.0f;
      #pragma unroll
      for (int i = 0; i < 8; ++i) {
        int hd = ej + 8 * i;
        float gr = lds_out[er][hd];
        float ir = lds_out[er][HID + hd];
        float dr = lds_out[er][2 * HID + hd];
        float g = fminf(fmaxf(sigmf(gr), 0.05f), 0.95f);
        c_reg[i] = (1.0f - g) * 0.9f * c_reg[i] + sigmf(ir) * tanhf(dr);
        v[i] = g * tanhf(c_reg[i]);
        sum += v[i]; sumsq += v[i] * v[i];
      }
      #pragma unroll
      for (int m = 1; m < 8; m <<= 1) {
        sum   += __shfl_xor(sum,   m, 32);
        sumsq += __shfl_xor(sumsq, m, 32);
      }
      float mean = sum * (1.0f / 64.0f);
      float var  = sumsq * (1.0f / 64.0f) - mean * mean;
      float rstd = rsqrtf(var + 1e-5f);
      float* orow = out + ((size_t)(row0 + er) * TSTEPS + t) * HID;
      #pragma unroll
      for (int i = 0; i < 8; ++i) {
        int hd = ej + 8 * i;
        float h = (v[i] - mean) * rstd * nrmw[i] + nrmb[i];
        lds_cat[er][INP + hd] = h;    // feeds next step's entry GEMM
        orow[hd] = h;
      }
    }
    __syncthreads();
  }

  // h_T = final h
  {
    float* ht = out + (size_t)Btotal * TSTEPS * HID + (size_t)(row0 + er) * HID;
    #pragma unroll
    for (int i = 0; i < 8; ++i) {
      int hd = ej + 8 * i;
      ht[hd] = lds_cat[er][INP + hd];
    }
  }
}

extern "C" void kernel_launch(void* const* d_in, const int* in_sizes, int n_in,
                              void* d_out, int out_size, void* d_ws, size_t ws_size,
                              hipStream_t stream) {
  const float* x        = (const float*)d_in[0];
  const float* fw       = (const float*)d_in[1];
  const float* entry_w  = (const float*)d_in[2];
  const float* entry_b  = (const float*)d_in[3];
  const float* exit_w   = (const float*)d_in[4];
  const float* exit_b   = (const float*)d_in[5];
  const float* vqc_w    = (const float*)d_in[6];
  const float* norm_w   = (const float*)d_in[7];
  const float* norm_b   = (const float*)d_in[8];
  float* out = (float*)d_out;

  int B = in_sizes[0] / (TSTEPS * INP);   // 256
  dim3 grid(B / ROWS), block(NTHR);
  cqssm_persistent_kernel<<<grid, block, 0, stream>>>(
      x, fw, entry_w, entry_b, exit_w, exit_b, vqc_w, norm_w, norm_b, out, B);
}